// LocalGrouper_88029649699401
// MI455X (gfx1250) — compile-verified
//
#include <hip/hip_runtime.h>
#include <float.h>
#include <math.h>

typedef float v2f __attribute__((ext_vector_type(2)));
typedef float v8f __attribute__((ext_vector_type(8)));

#define NB   8
#define NP   4096
#define NS   1024
#define KNN  32
#define NF   64
#define EPSV 1e-5f

// ---------------------------------------------------------------------------
// 1) Farthest point sampling: one 1024-thread workgroup per batch (32 wave32s,
//    fills a WGP). Each thread owns 4 points entirely in registers (coords +
//    running min-distance). Per step: 4 FMA chains -> 5-step shfl_xor wave
//    argmax -> 32-entry LDS reduce by wave 0. Only 2 barriers per step.
//    Records 'farthest' BEFORE the update (matches the jax scan collection).
// ---------------------------------------------------------------------------
__global__ void __launch_bounds__(1024)
fps_kernel(const float* __restrict__ xyz, int* __restrict__ fps_idx) {
    __shared__ float wv[32];
    __shared__ int   wi[32];
    __shared__ int   sFar;
    const int b    = blockIdx.x;
    const int tid  = threadIdx.x;        // 0..1023
    const int lane = tid & 31;
    const int wave = tid >> 5;           // 0..31
    const float* px = xyz + (size_t)b * 3 * NP;
    const float* py = px + NP;
    const float* pz = py + NP;

    float X[4], Y[4], Z[4], D[4];
    #pragma unroll
    for (int i = 0; i < 4; ++i) {
        const int n = tid + i * 1024;
        X[i] = px[n]; Y[i] = py[n]; Z[i] = pz[n];
        D[i] = 1e10f;
    }
    if (tid == 0) sFar = 0;
    __syncthreads();

    for (int it = 0; it < NS; ++it) {
        const int far = sFar;
        if (tid == 0) fps_idx[b * NS + it] = far;
        const float cx = px[far], cy = py[far], cz = pz[far];  // uniform, L2-hot
        float bestv = -1.0f; int besti = 0;
        #pragma unroll
        for (int i = 0; i < 4; ++i) {
            const float dx = X[i] - cx, dy = Y[i] - cy, dz = Z[i] - cz;
            float d = dx*dx + dy*dy + dz*dz;
            d = fminf(D[i], d);
            D[i] = d;
            if (d > bestv) { bestv = d; besti = tid + i * 1024; }  // strict > keeps lowest n
        }
        #pragma unroll
        for (int off = 16; off > 0; off >>= 1) {
            const float ov = __shfl_xor(bestv, off, 32);
            const int   oi = __shfl_xor(besti, off, 32);
            if (ov > bestv || (ov == bestv && oi < besti)) { bestv = ov; besti = oi; }
        }
        if (lane == 0) { wv[wave] = bestv; wi[wave] = besti; }
        __syncthreads();
        if (wave == 0) {
            float v = wv[lane]; int ii = wi[lane];
            #pragma unroll
            for (int off = 16; off > 0; off >>= 1) {
                const float ov = __shfl_xor(v, off, 32);
                const int   oi = __shfl_xor(ii, off, 32);
                if (ov > v || (ov == v && oi < ii)) { v = ov; ii = oi; }
            }
            if (lane == 0) sFar = ii;
        }
        __syncthreads();
    }
}

// ---------------------------------------------------------------------------
// 2) Gather sampled points/features; also emit transposed new_xyz / new_feats.
// ---------------------------------------------------------------------------
__global__ void gather_kernel(const float* __restrict__ xyz, const float* __restrict__ feats,
                              const int* __restrict__ fps_idx,
                              float* __restrict__ new_xyz_g, float* __restrict__ new_feats_g,
                              float* __restrict__ out) {
    const int gid = blockIdx.x * 256 + threadIdx.x;     // (b,s), 8192 total
    if (gid >= NB * NS) return;
    const int b = gid / NS, s = gid % NS;
    const int fi = fps_idx[gid];
    float* out_nx = out;                 // (B,3,S)
    float* out_nf = out + NB * 3 * NS;   // (B,64,S)
    for (int d = 0; d < 3; ++d) {
        const float v = xyz[((size_t)b * 3 + d) * NP + fi];
        new_xyz_g[(size_t)gid * 3 + d] = v;
        out_nx[((size_t)b * 3 + d) * NS + s] = v;
    }
    for (int c = 0; c < NF; ++c) {
        const float v = feats[((size_t)b * NF + c) * NP + fi];
        new_feats_g[(size_t)gid * NF + c] = v;
        out_nf[((size_t)b * NF + c) * NS + s] = v;
    }
}

__global__ void init_stats_kernel(float* __restrict__ stats) {
    if (threadIdx.x < 2 * NB) stats[threadIdx.x] = 0.0f;
}

// ---------------------------------------------------------------------------
// 3) kNN: 16 centroids per block (one WMMA M-tile), batch b = blockIdx.y.
//    Distances via V_WMMA_F32_16X16X4_F32:
//       A[m]   = (cx, cy, cz, |c|^2)       (16x4)
//       B[:,n] = (-2px, -2py, -2pz, 1)     (4x16)
//       C init = |p_n|^2 broadcast per column
//    Full 16x4096 f32 distance rows in 256 KB dynamic LDS (<= 320 KB/WGP).
//    Selection: wave w owns row w; each lane caches the min of its disjoint
//    stride-32 subset, so per round only the winner lane rescans (lazy rescan,
//    ~32x less LDS traffic; no cross-lane LDS visibility needed).
// ---------------------------------------------------------------------------
__global__ void knn_kernel(const float* __restrict__ xyz, const float* __restrict__ new_xyz_g,
                           int* __restrict__ knn_idx) {
    extern __shared__ float Dlds[];            // 16 * 4096 floats
    __shared__ float cA[16][4];
    const int tile = blockIdx.x;               // 0..63  (centroid tile)
    const int b    = blockIdx.y;
    const int tid  = threadIdx.x;              // 0..511 (16 waves)
    const int lane = tid & 31;
    const int wave = tid >> 5;
    const float* px = xyz + (size_t)b * 3 * NP;
    const float* py = px + NP;
    const float* pz = py + NP;

    if (tid < 16) {
        const int s = tile * 16 + tid;
        const float* c = new_xyz_g + ((size_t)b * NS + s) * 3;
        const float x = c[0], y = c[1], z = c[2];
        cA[tid][0] = x; cA[tid][1] = y; cA[tid][2] = z; cA[tid][3] = x*x + y*y + z*z;
    }
    __syncthreads();

    // A operand (16x4 f32 layout, ISA 7.12.2): lanes 0-15 -> K=0,1 ; lanes 16-31 -> K=2,3
    const int row = lane & 15;
    v2f a;
    if (lane < 16) { a.x = cA[row][0]; a.y = cA[row][1]; }
    else           { a.x = cA[row][2]; a.y = cA[row][3]; }

    for (int t = wave; t < NP / 16; t += 16) {       // each wave: 16 point-tiles
        const int n = t * 16 + row;
        const float x = px[n], y = py[n], z = pz[n];
        const float pn = x*x + y*y + z*z;            // |p|^2 for column 'row'
        v2f bb;                                      // B (4x16), mirrored layout
        if (lane < 16) { bb.x = -2.0f * x; bb.y = -2.0f * y; }
        else           { bb.x = -2.0f * z; bb.y = 1.0f; }
        v8f c;
        #pragma unroll
        for (int r = 0; r < 8; ++r) c[r] = pn;       // per-column |p|^2 accumulator init
        v8f d = __builtin_amdgcn_wmma_f32_16x16x4_f32(
            /*neg_a=*/false, a, /*neg_b=*/false, bb,
            /*c_mod=*/(short)0, c, /*reuse_a=*/false, /*reuse_b=*/false);
        const int mBase = (lane < 16) ? 0 : 8;       // C/D layout: VGPR r -> M=r / M=r+8
        #pragma unroll
        for (int r = 0; r < 8; ++r)
            Dlds[(size_t)(mBase + r) * NP + t * 16 + row] = d[r];
    }
    __syncthreads();

    // Selection with lazy per-lane rescan.
    float* rowD = Dlds + (size_t)wave * NP;
    const int sIdx    = tile * 16 + wave;
    const int outBase = (b * NS + sIdx) * KNN;

    float bv = FLT_MAX; int bi = NP;                 // cached min of this lane's subset
    for (int n = lane; n < NP; n += 32) {
        const float v = rowD[n];
        if (v < bv) { bv = v; bi = n; }              // strict < keeps lowest index
    }
    for (int k = 0; k < KNN; ++k) {
        float rv_ = bv; int ri_ = bi;                // cross-lane argmin (tie -> lowest idx)
        #pragma unroll
        for (int off = 16; off > 0; off >>= 1) {
            const float ov = __shfl_xor(rv_, off, 32);
            const int   oi = __shfl_xor(ri_, off, 32);
            if (ov < rv_ || (ov == rv_ && oi < ri_)) { rv_ = ov; ri_ = oi; }
        }
        if (lane == 0) knn_idx[outBase + k] = ri_;
        if (bi == ri_) {                             // exactly the owner lane (subsets disjoint)
            rowD[ri_] = FLT_MAX;                     // remove winner (same-lane write)
            bv = FLT_MAX; bi = NP;                   // rescan own 128 elements only
            for (int n = lane; n < NP; n += 32) {
                const float v = rowD[n];
                if (v < bv) { bv = v; bi = n; }
            }
        }
    }
}

// ---------------------------------------------------------------------------
// 4) Per-batch sum / sumsq of diff = grouped - new_feats (for std, ddof=1).
// ---------------------------------------------------------------------------
__global__ void stats_kernel(const float* __restrict__ feats, const float* __restrict__ new_feats_g,
                             const int* __restrict__ knn_idx, float* __restrict__ stats) {
    __shared__ float sv[256], sq[256];
    const int gid = blockIdx.x;                 // (b,s)
    const int b   = gid >> 10;
    const int tid = threadIdx.x;
    const int*   kix = knn_idx     + (size_t)gid * KNN;
    const float* nf  = new_feats_g + (size_t)gid * NF;
    float s1 = 0.0f, s2 = 0.0f;
    for (int e = tid; e < KNN * NF; e += 256) {
        const int j = e >> 6, c = e & 63;
        const int pt = kix[j];
        const float v = feats[((size_t)b * NF + c) * NP + pt] - nf[c];
        s1 += v; s2 += v * v;
    }
    sv[tid] = s1; sq[tid] = s2;
    __syncthreads();
    for (int s = 128; s > 0; s >>= 1) {
        if (tid < s) { sv[tid] += sv[tid + s]; sq[tid] += sq[tid + s]; }
        __syncthreads();
    }
    if (tid == 0) {
        atomicAdd(&stats[b * 2 + 0], sv[0]);
        atomicAdd(&stats[b * 2 + 1], sq[0]);
    }
}

// ---------------------------------------------------------------------------
// 5) Finalize: normalize, affine, concat, write (B,128,S,32) output.
//    134 MB of stores, coalesced along the inner k=32 dimension ->
//    bandwidth-bound against the 23.3 TB/s HBM roofline.
// ---------------------------------------------------------------------------
__global__ void finalize_kernel(const float* __restrict__ feats, const float* __restrict__ new_feats_g,
                                const int* __restrict__ knn_idx,
                                const float* __restrict__ alpha, const float* __restrict__ beta,
                                const float* __restrict__ stats, float* __restrict__ out) {
    const int gid = blockIdx.x * 256 + threadIdx.x;   // (b,s,j): 262144 total
    if (gid >= NB * NS * KNN) return;
    const int j = gid & (KNN - 1);
    const int s = (gid >> 5) & (NS - 1);
    const int b = gid >> 15;
    const float n = (float)(NS * KNN * NF);
    const float s1 = stats[b * 2 + 0], s2 = stats[b * 2 + 1];
    const float mean = s1 / n;
    const float var  = (s2 - s1 * mean) / (n - 1.0f);
    const float stdv = sqrtf(fmaxf(var, 0.0f));
    const float inv  = 1.0f / (stdv + EPSV);
    const int pt = knn_idx[gid];
    const float* nf = new_feats_g + ((size_t)b * NS + s) * NF;
    float* pf = out + NB * 3 * NS + NB * NF * NS;     // (B,128,S,K)
    for (int c = 0; c < NF; ++c) {
        const float v   = feats[((size_t)b * NF + c) * NP + pt];
        const float nfc = nf[c];
        const float g   = alpha[c] * ((v - nfc) * inv) + beta[c];
        pf[(((size_t)b * 128 + c)      * NS + s) * KNN + j] = g;
        pf[(((size_t)b * 128 + c + NF) * NS + s) * KNN + j] = g - nfc;
    }
}

// ---------------------------------------------------------------------------
extern "C" void kernel_launch(void* const* d_in, const int* in_sizes, int n_in,
                              void* d_out, int out_size, void* d_ws, size_t ws_size,
                              hipStream_t stream) {
    (void)in_sizes; (void)n_in; (void)out_size; (void)ws_size;
    const float* xyz   = (const float*)d_in[0];
    const float* feats = (const float*)d_in[1];
    const float* alpha = (const float*)d_in[2];
    const float* beta  = (const float*)d_in[3];
    float* out = (float*)d_out;

    char* ws = (char*)d_ws;
    int*   fps_idx     = (int*)(ws);                                        // 32 KB
    int*   knn_idx     = (int*)(ws + 32768);                                // 1 MB
    float* new_xyz_g   = (float*)(ws + 32768 + 1048576);                    // 96 KB
    float* new_feats_g = (float*)(ws + 32768 + 1048576 + 98304);            // 2 MB
    float* stats       = (float*)(ws + 32768 + 1048576 + 98304 + 2097152);  // 64 B

    const int knn_lds = 16 * NP * (int)sizeof(float);                       // 256 KB
    (void)hipFuncSetAttribute(reinterpret_cast<const void*>(knn_kernel),
                              hipFuncAttributeMaxDynamicSharedMemorySize, knn_lds);

    fps_kernel<<<NB, 1024, 0, stream>>>(xyz, fps_idx);
    gather_kernel<<<(NB * NS) / 256, 256, 0, stream>>>(xyz, feats, fps_idx,
                                                       new_xyz_g, new_feats_g, out);
    init_stats_kernel<<<1, 32, 0, stream>>>(stats);
    knn_kernel<<<dim3(NS / 16, NB), 512, knn_lds, stream>>>(xyz, new_xyz_g, knn_idx);
    stats_kernel<<<NB * NS, 256, 0, stream>>>(feats, new_feats_g, knn_idx, stats);
    finalize_kernel<<<(NB * NS * KNN) / 256, 256, 0, stream>>>(feats, new_feats_g, knn_idx,
                                                               alpha, beta, stats, out);
}